// InterResidueVDWRepulsiveLoss_80736795230591
// MI455X (gfx1250) — compile-verified
//
#include <hip/hip_runtime.h>
#include <hip/hip_bf16.h>

typedef float v2f __attribute__((ext_vector_type(2)));
typedef float v8f __attribute__((ext_vector_type(8)));

#define NRES   256
#define NATOM  37
#define NTOT   (NRES * NATOM)      /* 9472 */
#define NTILE  (NTOT / 16)         /* 592 column tiles */
#define ROWS_PER_BLOCK 32
#define NBLK   (NTOT / ROWS_PER_BLOCK)  /* 296 row-strip blocks */
#define WAVES_PER_BLOCK 8

// clamp lower bound = safe_norm eps (3*1e-8); upper bound finite so LLVM
// cannot fold the med3 into canonicalize+max
#define D2_LO 3.0e-8f
#define D2_HI 3.0e38f

// ---------------------------------------------------------------------------
// Prep: per-atom attributes, computed once.
//   bneg[j] = {-2x, -2y, -2z, 0}   (B fragment loads directly as b64 halves)
//   meta[j] = {|x|^2, rad_masked, res_as_float, 0}
// ---------------------------------------------------------------------------
__global__ __launch_bounds__(256)
void vdw_prep_kernel(const float* __restrict__ coords,
                     const float* __restrict__ vdw,
                     const int*   __restrict__ mask,
                     float4* __restrict__ bneg,
                     float4* __restrict__ meta)
{
    const int j = blockIdx.x * blockDim.x + threadIdx.x;
    if (j >= NTOT) return;
    const float x0 = coords[j * 3 + 0];
    const float x1 = coords[j * 3 + 1];
    const float x2 = coords[j * 3 + 2];
    const float sq = fmaf(x0, x0, fmaf(x1, x1, x2 * x2));
    bneg[j] = make_float4(-2.0f * x0, -2.0f * x1, -2.0f * x2, 0.0f);
    const float rad = mask[j] ? vdw[(unsigned)j % 37u] : -1e9f;
    meta[j] = make_float4(sq, rad, (float)(j / 37), 0.0f);
}

// ---------------------------------------------------------------------------
// One block owns a 32-row strip (two 16-row tiles); 8 waves sweep the 592
// column tiles, two v_wmma_f32_16x16x4_f32 per column tile:
//   D = A(16x4) * B(4x16) + C,  A=x_i (K padded), B=-2 x_j, C=|xi|^2+|xj|^2
// Residue-separation mask is applied only on the ~2% of tiles that intersect
// the |res_i - res_j| <= 1 band (wave-uniform scalar branch).
// ---------------------------------------------------------------------------
__global__ __launch_bounds__(256)
void vdw_tiles_kernel(const float*  __restrict__ coords,
                      const float*  __restrict__ bnegf,  // as flat floats
                      const float4* __restrict__ meta,
                      float* __restrict__ blk_sum,
                      float* __restrict__ blk_cnt)
{
    __shared__ float s_x0[ROWS_PER_BLOCK], s_x1[ROWS_PER_BLOCK], s_x2[ROWS_PER_BLOCK];
    __shared__ float s_sq[ROWS_PER_BLOCK], s_rad[ROWS_PER_BLOCK], s_resF[ROWS_PER_BLOCK];
    __shared__ float s_wsum[WAVES_PER_BLOCK * ROWS_PER_BLOCK];

    const int tid = threadIdx.x;
    const int i0  = blockIdx.x * ROWS_PER_BLOCK;

    if (tid < ROWS_PER_BLOCK) {
        const int gi = i0 + tid;
        s_x0[tid] = coords[gi * 3 + 0];
        s_x1[tid] = coords[gi * 3 + 1];
        s_x2[tid] = coords[gi * 3 + 2];
        const float4 mt = meta[gi];
        s_sq[tid] = mt.x; s_rad[tid] = mt.y; s_resF[tid] = mt.z;
    }
    __syncthreads();

    const int lane = tid & 31;
    const int wid  = __builtin_amdgcn_readfirstlane(tid >> 5);  // SGPR wave id
    const int hi   = lane >> 4;     // 0: rows 0-7 / K 0-1, 1: rows 8-15 / K 2-3
    const int m    = lane & 15;     // column within tile

    // residue range of this block's 32 rows (scalar)
    const int reshi_p1 = (i0 + ROWS_PER_BLOCK - 1) / 37 + 1;
    const int reslo_m1 = i0 / 37 - 1;

    // A fragments for the two 16-row tiles
    v2f afrag[2];
#pragma unroll
    for (int t = 0; t < 2; ++t) {
        const int r = t * 16 + m;
        afrag[t].x = hi ? s_x2[r] : s_x0[r];
        afrag[t].y = hi ? 0.0f    : s_x1[r];
    }

    // per-half row attributes for both tiles
    float sqi[2][8], radi[2][8], resiF[2][8];
#pragma unroll
    for (int t = 0; t < 2; ++t)
#pragma unroll
        for (int v = 0; v < 8; ++v) {
            const int r = t * 16 + hi * 8 + v;
            sqi[t][v]   = s_sq[r];
            radi[t][v]  = s_rad[r] + 0.25f;     // fold TOL into row radius
            resiF[t][v] = s_resF[r];
        }

    float racc[2][8];
#pragma unroll
    for (int t = 0; t < 2; ++t)
#pragma unroll
        for (int v = 0; v < 8; ++v) racc[t][v] = 0.0f;

    for (int jt = wid; jt < NTILE; jt += WAVES_PER_BLOCK) {
        const int col = jt * 16 + m;
        // B fragment: direct b64 load of this lane's {K pair} from bneg
        const v2f bfrag = *(const v2f*)(bnegf + col * 4 + hi * 2);
        const float4 mt = meta[col];
        const float sqj  = mt.x;
        const float radj = mt.y;
        const float rjF  = mt.z;

        v8f d[2];
#pragma unroll
        for (int t = 0; t < 2; ++t) {
            v8f c;
#pragma unroll
            for (int v = 0; v < 8; ++v) c[v] = sqi[t][v] + sqj;
            d[t] = __builtin_amdgcn_wmma_f32_16x16x4_f32(
                false, afrag[t], false, bfrag, (short)0, c, false, false);
        }

        // wave-uniform: does this column tile intersect the banned band?
        const int j0   = jt * 16;
        const int rjlo = j0 / 37;
        const int rjhi = (j0 + 15) / 37;
        const bool clean = (rjlo > reshi_p1) || (rjhi < reslo_m1);

        if (clean) {
#pragma unroll
            for (int t = 0; t < 2; ++t)
#pragma unroll
                for (int v = 0; v < 8; ++v) {
                    const float d2   = __builtin_amdgcn_fmed3f(d[t][v], D2_LO, D2_HI);
                    const float dist = __builtin_amdgcn_sqrtf(d2);
                    const float viol = fmaxf((radi[t][v] + radj) - dist, 0.0f);
                    racc[t][v] = fmaf(viol, viol, racc[t][v]);
                }
        } else {
#pragma unroll
            for (int t = 0; t < 2; ++t)
#pragma unroll
                for (int v = 0; v < 8; ++v) {
                    const float d2   = __builtin_amdgcn_fmed3f(d[t][v], D2_LO, D2_HI);
                    const float dist = __builtin_amdgcn_sqrtf(d2);
                    const float viol = fmaxf((radi[t][v] + radj) - dist, 0.0f);
                    const bool  sep  = fabsf(resiF[t][v] - rjF) > 1.5f;
                    const float w    = sep ? viol : 0.0f;
                    racc[t][v] = fmaf(w, w, racc[t][v]);
                }
        }
    }

    // reduce each row accumulator across the 16 lanes of its half-group
#pragma unroll
    for (int t = 0; t < 2; ++t)
#pragma unroll
        for (int v = 0; v < 8; ++v) {
#pragma unroll
            for (int s = 1; s < 16; s <<= 1)
                racc[t][v] += __shfl_xor(racc[t][v], s, 32);
        }
    if (m == 0) {
#pragma unroll
        for (int t = 0; t < 2; ++t)
#pragma unroll
            for (int v = 0; v < 8; ++v)
                s_wsum[wid * ROWS_PER_BLOCK + t * 16 + hi * 8 + v] = racc[t][v];
    }
    __syncthreads();

    // deterministic per-block reduction (fixed wave order)
    if (tid == 0) {
        float bs = 0.0f, bc = 0.0f;
        for (int r = 0; r < ROWS_PER_BLOCK; ++r) {
            float tot = 0.0f;
            for (int w = 0; w < WAVES_PER_BLOCK; ++w)
                tot += s_wsum[w * ROWS_PER_BLOCK + r];
            tot *= 0.5f;                       // viols = row_sum / 2
            bs += tot;
            if (tot > 0.0f) bc += 1.0f;
        }
        blk_sum[blockIdx.x] = bs;
        blk_cnt[blockIdx.x] = bc;
    }
}

__global__ void vdw_finalize_kernel(const float* __restrict__ blk_sum,
                                    const float* __restrict__ blk_cnt,
                                    float* __restrict__ out)
{
    if (threadIdx.x == 0 && blockIdx.x == 0) {
        float s = 0.0f, c = 0.0f;
        for (int i = 0; i < NBLK; ++i) { s += blk_sum[i]; c += blk_cnt[i]; }
        out[0] = s / fmaxf(c, 1.0f);
    }
}

extern "C" void kernel_launch(void* const* d_in, const int* in_sizes, int n_in,
                              void* d_out, int out_size, void* d_ws, size_t ws_size,
                              hipStream_t stream) {
    const float* coords = (const float*)d_in[0];   // (1,256,37,3) f32
    const float* vdw    = (const float*)d_in[1];   // (37,) f32
    const int*   mask   = (const int*)d_in[2];     // (1,256,37) int
    float* out = (float*)d_out;

    // workspace layout
    char* ws = (char*)d_ws;
    float4* bneg = (float4*)ws;                    ws += NTOT * sizeof(float4);
    float4* meta = (float4*)ws;                    ws += NTOT * sizeof(float4);
    float*  blk_sum = (float*)ws;                  ws += NBLK * sizeof(float);
    float*  blk_cnt = (float*)ws;

    vdw_prep_kernel<<<(NTOT + 255) / 256, 256, 0, stream>>>(coords, vdw, mask,
                                                            bneg, meta);
    vdw_tiles_kernel<<<NBLK, 256, 0, stream>>>(coords, (const float*)bneg, meta,
                                               blk_sum, blk_cnt);
    vdw_finalize_kernel<<<1, 32, 0, stream>>>(blk_sum, blk_cnt, out);
}